// TrueHGCN_23691039604878
// MI455X (gfx1250) — compile-verified
//
#include <hip/hip_runtime.h>
#include <math.h>

namespace {

constexpr int G_    = 128;
constexpr int NPER_ = 512;
constexpr int NTOT_ = G_ * NPER_;   // 65536
constexpr int ETOT_ = NTOT_ * 16;   // 1048576
constexpr int HID_  = 128;
constexpr int INC_  = 100;
constexpr float EPS_ = 1e-5f;

typedef __attribute__((ext_vector_type(2))) float v2f;
typedef __attribute__((ext_vector_type(8))) float v8f;

// ---------------------------------------------------------------- utilities
__global__ void zero_kernel(float* __restrict__ p, int n) {
  int i = blockIdx.x * blockDim.x + threadIdx.x;
  if (i < n) p[i] = 0.0f;
}

__global__ void init_deg_kernel(float* __restrict__ deg, int n) {
  int i = blockIdx.x * blockDim.x + threadIdx.x;
  if (i < n) deg[i] = 1.0f;   // self-loop contributes 1
}

__global__ void accum_deg_kernel(const int* __restrict__ dst,
                                 const int* __restrict__ mask,
                                 float* __restrict__ deg, int e) {
  int i = blockIdx.x * blockDim.x + threadIdx.x;
  if (i >= e) return;
  if (mask && !mask[i]) return;
  atomicAdd(&deg[dst[i]], 1.0f);
}

// ------------------------------------------------- fp32 WMMA GEMM (16x16x4)
// C[M,N] = A[M,K] @ B[K,N] (+bias, optional relu). M mult of 16, N mult of 64,
// K mult of 4 (and even). One wave computes a 16x64 strip (4 accumulators) so
// each A fragment (global_load_b64) feeds 4 WMMAs; B is staged transposed in
// LDS so each B fragment is one aligned ds_load_b64.
__global__ void wmma_gemm_kernel(const float* __restrict__ A,
                                 const float* __restrict__ B,
                                 const float* __restrict__ bias,
                                 float* __restrict__ C,
                                 int M, int K, int N, int relu, int useLds) {
  extern __shared__ float BsT[];          // [N][K] (transposed)
  if (useLds) {
    int tot = K * N;
    for (int i = threadIdx.x; i < tot; i += blockDim.x) {
      int k = i / N, n = i - k * N;       // coalesced global read
      BsT[n * K + k] = B[i];
    }
  }
  __syncthreads();

  const int lane = threadIdx.x & 31;
  const int wave = threadIdx.x >> 5;
  const int strips = N >> 6;              // strips of 64 columns
  const int ntM = M >> 4;
  const int tile = blockIdx.x * (blockDim.x >> 5) + wave;
  if (tile >= ntM * strips) return;       // wave-uniform exit (EXEC stays full)
  const int mt = tile / strips;
  const int st = tile - mt * strips;
  const int l15 = lane & 15;
  const int ksel = (lane >> 4) << 1;      // lanes 0-15 -> K {0,1}; 16-31 -> {2,3}
  const float* Arow = A + (size_t)(mt * 16 + l15) * K;
  const int n0 = st * 64 + l15;

  v8f acc0 = {}, acc1 = {}, acc2 = {}, acc3 = {};
  if (useLds) {
    const int o0 = (n0 +  0) * K;
    const int o1 = (n0 + 16) * K;
    const int o2 = (n0 + 32) * K;
    const int o3 = (n0 + 48) * K;
    for (int k = 0; k < K; k += 4) {
      int kk = k + ksel;
      v2f a  = *(const v2f*)(Arow + kk);          // global_load_b64
      v2f b0 = *(const v2f*)(&BsT[o0 + kk]);      // ds_load_b64
      v2f b1 = *(const v2f*)(&BsT[o1 + kk]);
      v2f b2 = *(const v2f*)(&BsT[o2 + kk]);
      v2f b3 = *(const v2f*)(&BsT[o3 + kk]);
      acc0 = __builtin_amdgcn_wmma_f32_16x16x4_f32(false, a, false, b0, (short)0, acc0, false, false);
      acc1 = __builtin_amdgcn_wmma_f32_16x16x4_f32(false, a, false, b1, (short)0, acc1, false, false);
      acc2 = __builtin_amdgcn_wmma_f32_16x16x4_f32(false, a, false, b2, (short)0, acc2, false, false);
      acc3 = __builtin_amdgcn_wmma_f32_16x16x4_f32(false, a, false, b3, (short)0, acc3, false, false);
    }
  } else {
    for (int k = 0; k < K; k += 4) {
      int kk = k + ksel;
      v2f a = *(const v2f*)(Arow + kk);
      const float* Br0 = B + (size_t)kk * N;
      const float* Br1 = Br0 + N;
      v2f b0 = {Br0[n0 +  0], Br1[n0 +  0]};
      v2f b1 = {Br0[n0 + 16], Br1[n0 + 16]};
      v2f b2 = {Br0[n0 + 32], Br1[n0 + 32]};
      v2f b3 = {Br0[n0 + 48], Br1[n0 + 48]};
      acc0 = __builtin_amdgcn_wmma_f32_16x16x4_f32(false, a, false, b0, (short)0, acc0, false, false);
      acc1 = __builtin_amdgcn_wmma_f32_16x16x4_f32(false, a, false, b1, (short)0, acc1, false, false);
      acc2 = __builtin_amdgcn_wmma_f32_16x16x4_f32(false, a, false, b2, (short)0, acc2, false, false);
      acc3 = __builtin_amdgcn_wmma_f32_16x16x4_f32(false, a, false, b3, (short)0, acc3, false, false);
    }
  }

  const int half = lane >> 4;             // C: VGPR v -> row v (+8 for hi lanes)
  v8f accs[4] = {acc0, acc1, acc2, acc3};
#pragma unroll
  for (int t = 0; t < 4; ++t) {
    int col = st * 64 + t * 16 + l15;
    float bv = bias ? bias[col] : 0.0f;
#pragma unroll
    for (int v = 0; v < 8; ++v) {
      int row = mt * 16 + v + half * 8;
      float val = accs[t][v] + bv;
      if (relu) val = fmaxf(val, 0.0f);
      C[(size_t)row * N + col] = val;
    }
  }
}

// ------------------------------------------------- edge scatter (GCN message)
// 128 lanes cover the channel dim of one edge; atomics land in L2-resident agg.
__global__ void scatter_kernel(const int* __restrict__ src,
                               const int* __restrict__ dst,
                               const int* __restrict__ mask,
                               const float* __restrict__ deg,
                               const float* __restrict__ xl,
                               float* __restrict__ agg, int e) {
  int edge = blockIdx.x * 2 + (threadIdx.x >> 7);
  int ch = threadIdx.x & 127;
  if (edge >= e) return;
  if (mask && !mask[edge]) return;
  int s = src[edge], d = dst[edge];
  float coef = rsqrtf(deg[s]) * rsqrtf(deg[d]);
  atomicAdd(&agg[(size_t)d * HID_ + ch], coef * xl[(size_t)s * HID_ + ch]);
}

// pre = agg + (1/deg)*xl + b  (in place into agg) + per-channel BN partials
__global__ void finalize_stats_kernel(float* __restrict__ agg,
                                      const float* __restrict__ xl,
                                      const float* __restrict__ deg,
                                      const float* __restrict__ b,
                                      float* __restrict__ sum,
                                      float* __restrict__ sumsq, int M) {
  int ch = threadIdx.x;                   // 128 threads, one per channel
  int row0 = blockIdx.x * 64;
  float s = 0.0f, s2 = 0.0f;
  float bv = b[ch];
  for (int r = 0; r < 64; ++r) {
    int row = row0 + r;
    if (row >= M) break;
    float dinv2 = 1.0f / deg[row];
    size_t idx = (size_t)row * HID_ + ch;
    float v = agg[idx] + dinv2 * xl[idx] + bv;
    agg[idx] = v;
    s += v; s2 += v * v;
  }
  atomicAdd(&sum[ch], s);
  atomicAdd(&sumsq[ch], s2);
}

__global__ void bn_relu_kernel(float* __restrict__ h,
                               const float* __restrict__ sum,
                               const float* __restrict__ sumsq,
                               const float* __restrict__ gamma,
                               const float* __restrict__ beta, int M) {
  int i = blockIdx.x * blockDim.x + threadIdx.x;
  if (i >= M * HID_) return;
  int ch = i & (HID_ - 1);
  float inv_m = 1.0f / (float)M;
  float mu = sum[ch] * inv_m;
  float var = sumsq[ch] * inv_m - mu * mu;   // biased (ddof=0), matches jnp.var
  float v = (h[i] - mu) * rsqrtf(var + EPS_) * gamma[ch] + beta[ch];
  h[i] = fmaxf(v, 0.0f);
}

__global__ void attnorm_kernel(const float* __restrict__ att,
                               float* __restrict__ invn) {
  __shared__ float red[128];
  float v = att[threadIdx.x];
  red[threadIdx.x] = v * v;
  __syncthreads();
  for (int s = 64; s > 0; s >>= 1) {
    if (threadIdx.x < s) red[threadIdx.x] += red[threadIdx.x + s];
    __syncthreads();
  }
  if (threadIdx.x == 0) invn[0] = rsqrtf(red[0]);
}

__global__ void score_kernel(const float* __restrict__ h,
                             const float* __restrict__ att,
                             const float* __restrict__ invn,
                             float* __restrict__ score, int M) {
  int i = blockIdx.x * blockDim.x + threadIdx.x;
  if (i >= M) return;
  const float* row = h + (size_t)i * HID_;
  float s = 0.0f;
  for (int c = 0; c < HID_; ++c) s += row[c] * att[c];
  score[i] = tanhf(s * invn[0]);
}

// one block per graph: rank each node (desc score, index tie-break == lax.top_k
// ordering), keep rank<k, compact by rank, gate features by score.
__global__ void topk_kernel(const float* __restrict__ h,
                            const float* __restrict__ score,
                            float* __restrict__ hout,
                            int* __restrict__ old2new, int n, int k) {
  extern __shared__ float ssc[];
  int g = blockIdx.x;
  const float* sc = score + (size_t)g * n;
  for (int i = threadIdx.x; i < n; i += blockDim.x) ssc[i] = sc[i];
  __syncthreads();
  for (int i = threadIdx.x; i < n; i += blockDim.x) {
    float si = ssc[i];
    int rank = 0;
    for (int j = 0; j < n; ++j) {
      float sj = ssc[j];
      rank += (sj > si) || (sj == si && j < i);
    }
    int oldg = g * n + i;
    if (rank < k) {
      int newg = g * k + rank;
      old2new[oldg] = newg;
      const float* src = h + (size_t)oldg * HID_;
      float* dst = hout + (size_t)newg * HID_;
      for (int c = 0; c < HID_; ++c) dst[c] = src[c] * si;
    } else {
      old2new[oldg] = -1;
    }
  }
}

__global__ void remap_kernel(const int* __restrict__ srcIn,
                             const int* __restrict__ dstIn,
                             const int* __restrict__ maskIn,
                             const int* __restrict__ old2new,
                             int* __restrict__ srcOut,
                             int* __restrict__ dstOut,
                             int* __restrict__ maskOut, int e) {
  int i = blockIdx.x * blockDim.x + threadIdx.x;
  if (i >= e) return;
  int m = maskIn ? maskIn[i] : 1;
  int s2 = old2new[srcIn[i]];
  int d2 = old2new[dstIn[i]];
  int valid = m && (s2 >= 0) && (d2 >= 0);
  srcOut[i] = s2 >= 0 ? s2 : 0;
  dstOut[i] = d2 >= 0 ? d2 : 0;
  maskOut[i] = valid;
}

// accumulate [max-pool | mean-pool] per graph into sread (x1+x2+x3 fused)
__global__ void readout_kernel(const float* __restrict__ h,
                               float* __restrict__ sread, int k) {
  int g = blockIdx.x;
  int ch = threadIdx.x;                   // 128
  const float* base = h + ((size_t)g * k) * HID_ + ch;
  float mx = -3.402823466e38f, sm = 0.0f;
  for (int r = 0; r < k; ++r) {
    float v = base[(size_t)r * HID_];
    mx = fmaxf(mx, v);
    sm += v;
  }
  sread[g * 256 + ch] += mx;
  sread[g * 256 + 128 + ch] += sm / (float)k;
}

__global__ void fc2_kernel(const float* __restrict__ s,
                           const float* __restrict__ W,
                           const float* __restrict__ b,
                           float* __restrict__ out) {
  int t = blockIdx.x * blockDim.x + threadIdx.x;   // 256 = G*2
  if (t >= G_ * 2) return;
  int g = t >> 1, o = t & 1;
  float acc = b[o];
  const float* row = s + (size_t)g * HID_;
  for (int c = 0; c < HID_; ++c) acc += row[c] * W[c * 2 + o];
  out[t] = acc;
}

}  // namespace

extern "C" void kernel_launch(void* const* d_in, const int* in_sizes, int n_in,
                              void* d_out, int out_size, void* d_ws, size_t ws_size,
                              hipStream_t stream) {
  (void)in_sizes; (void)n_in; (void)out_size; (void)ws_size;
  const float* x  = (const float*)d_in[0];
  const int*   ei = (const int*)d_in[1];
  const float* Wp = (const float*)d_in[3];
  const float* bp = (const float*)d_in[4];
  const float* Wc[3]  = {(const float*)d_in[5],  (const float*)d_in[10], (const float*)d_in[15]};
  const float* bc[3]  = {(const float*)d_in[6],  (const float*)d_in[11], (const float*)d_in[16]};
  const float* gc[3]  = {(const float*)d_in[7],  (const float*)d_in[12], (const float*)d_in[17]};
  const float* bec[3] = {(const float*)d_in[8],  (const float*)d_in[13], (const float*)d_in[18]};
  const float* atc[3] = {(const float*)d_in[9],  (const float*)d_in[14], (const float*)d_in[19]};
  const float* Wl1 = (const float*)d_in[20];
  const float* bl1 = (const float*)d_in[21];
  const float* Wl2 = (const float*)d_in[22];
  const float* bl2 = (const float*)d_in[23];

  char* ws = (char*)d_ws;
  size_t off = 0;
  auto alloc = [&](size_t bytes) -> char* {
    char* p = ws + off;
    off += (bytes + 255) & ~(size_t)255;
    return p;
  };
  float* bufA  = (float*)alloc((size_t)NTOT_ * HID_ * 4);
  float* bufB  = (float*)alloc((size_t)NTOT_ * HID_ * 4);
  float* bufC  = (float*)alloc((size_t)NTOT_ * HID_ * 4);
  float* deg   = (float*)alloc((size_t)NTOT_ * 4);
  float* score = (float*)alloc((size_t)NTOT_ * 4);
  float* sread = (float*)alloc((size_t)G_ * 256 * 4);
  float* sfc1  = (float*)alloc((size_t)G_ * HID_ * 4);
  float* stats = (float*)alloc(1024 * 4);           // sum[128], sumsq[128], invn
  int*   o2n   = (int*)alloc((size_t)NTOT_ * 4);
  int*   eA    = (int*)alloc((size_t)3 * ETOT_ * 4);
  int*   eB    = (int*)alloc((size_t)3 * ETOT_ * 4);

  auto gemm = [&](const float* A, const float* B, const float* bias, float* C,
                  int M, int K, int N, int relu, int useLds) {
    int tiles = (M / 16) * (N / 64);      // 16x64 strip per wave
    int blocks = (tiles + 7) / 8;
    size_t sh = useLds ? (size_t)K * N * 4 : 0;
    wmma_gemm_kernel<<<blocks, 256, sh, stream>>>(A, B, bias, C, M, K, N, relu, useLds);
  };

  // h0 = relu(x @ Wp + bp)
  gemm(x, Wp, bp, bufA, NTOT_, INC_, HID_, 1, 1);
  zero_kernel<<<(G_ * 256 + 255) / 256, 256, 0, stream>>>(sread, G_ * 256);

  float* pH = bufA;   // layer input features
  float* pXL = bufB;  // linear output / pooled output
  float* pAGG = bufC; // aggregation / post-BN features

  const int Ms[3]   = {NTOT_, G_ * 410, G_ * 328};   // conv input node counts
  const int nper[3] = {512, 410, 328};
  const int kk[3]   = {410, 328, 263};
  const int* srcCur = ei;
  const int* dstCur = ei + ETOT_;
  const int* mCur = nullptr;
  int* epool[2] = {eA, eB};

  for (int L = 0; L < 3; ++L) {
    int M = Ms[L];
    // degree (self-loop + masked in-edges)
    init_deg_kernel<<<(M + 255) / 256, 256, 0, stream>>>(deg, M);
    accum_deg_kernel<<<(ETOT_ + 255) / 256, 256, 0, stream>>>(dstCur, mCur, deg, ETOT_);
    // xl = h @ W
    gemm(pH, Wc[L], nullptr, pXL, M, HID_, HID_, 0, 1);
    // agg = scatter_add(coef * xl[src] -> dst)
    zero_kernel<<<(M * HID_ + 255) / 256, 256, 0, stream>>>(pAGG, M * HID_);
    scatter_kernel<<<ETOT_ / 2, 256, 0, stream>>>(srcCur, dstCur, mCur, deg, pXL, pAGG, ETOT_);
    // pre = agg + dinv^2*xl + b ; BN stats
    zero_kernel<<<1, 256, 0, stream>>>(stats, 256);
    finalize_stats_kernel<<<(M + 63) / 64, 128, 0, stream>>>(pAGG, pXL, deg, bc[L],
                                                             stats, stats + 128, M);
    bn_relu_kernel<<<(M * HID_ + 255) / 256, 256, 0, stream>>>(pAGG, stats, stats + 128,
                                                               gc[L], bec[L], M);
    // scores + top-k pooling (pooled features -> pXL, which is now free)
    attnorm_kernel<<<1, 128, 0, stream>>>(atc[L], stats + 256);
    score_kernel<<<(M + 255) / 256, 256, 0, stream>>>(pAGG, atc[L], stats + 256, score, M);
    topk_kernel<<<G_, 512, (size_t)nper[L] * 4, stream>>>(pAGG, score, pXL, o2n,
                                                          nper[L], kk[L]);
    readout_kernel<<<G_, 128, 0, stream>>>(pXL, sread, kk[L]);
    if (L < 2) {
      int* eo = epool[L];
      remap_kernel<<<(ETOT_ + 255) / 256, 256, 0, stream>>>(srcCur, dstCur, mCur, o2n,
                                                            eo, eo + ETOT_, eo + 2 * ETOT_,
                                                            ETOT_);
      srcCur = eo; dstCur = eo + ETOT_; mCur = eo + 2 * ETOT_;
    }
    // pooled features become next layer's input
    float* t = pH; pH = pXL; pXL = t;
  }

  // MLP head: relu(sread @ Wl1 + bl1) @ Wl2 + bl2
  gemm(sread, Wl1, bl1, sfc1, G_, 2 * HID_, HID_, 1, 0);
  fc2_kernel<<<1, 256, 0, stream>>>(sfc1, Wl2, bl2, (float*)d_out);
}